// Embedding_46273977647569
// MI455X (gfx1250) — compile-verified
//
#include <hip/hip_runtime.h>

// Embedding gather: out[t, :] = weight[ids[t], :]
// B*S = 8192 tokens, D = 1024 fp32 -> 4 KB per row.
// Pure HBM-bound: ~64 MB traffic -> ~2.7 us at 23.3 TB/s.
//
// Per block (256 threads, 8 wave32): 1x s_load_b128 (4 token ids),
// 4x global_load_b128 (independent, 4-deep MLP per thread),
// 4x global_store_b128 NT at base + {0,4096,8192,12288} immediate offsets.

typedef float v4f __attribute__((ext_vector_type(4)));

constexpr int D_DIM = 1024;   // fp32 elements per row (fixed by reference)
constexpr int TPB   = 4;      // tokens per block

__global__ __launch_bounds__(256) void embed_gather_kernel(
    const int* __restrict__ ids,
    const float* __restrict__ weight,
    float* __restrict__ out,
    int n_tokens)
{
    const unsigned tid = threadIdx.x;            // 0..255 -> float4 chunk index
    __builtin_assume(tid < 256);
    const int t0 = blockIdx.x * TPB;
    const unsigned off = tid * 4u;               // float offset of this lane's 16B chunk

    if (t0 + TPB <= n_tokens) {
        // ---- fast path: straight-line, no branches ----
        unsigned rows[TPB];
#pragma unroll
        for (int k = 0; k < TPB; ++k)
            rows[k] = (unsigned)ids[t0 + k];     // one s_load_b128 (uniform, consecutive)

        v4f v[TPB];
#pragma unroll
        for (int k = 0; k < TPB; ++k)            // 4 independent b128 loads in flight
            v[k] = *reinterpret_cast<const v4f*>(weight + (size_t)rows[k] * D_DIM + off);

        // Single uniform store base; k*D folds into the 24-bit instruction offset.
        float* dstbase = out + (size_t)t0 * D_DIM + off;
#pragma unroll
        for (int k = 0; k < TPB; ++k)
            __builtin_nontemporal_store(
                v[k], reinterpret_cast<v4f*>(dstbase + k * D_DIM));
    } else {
        // ---- ragged tail (never taken for 8192 tokens) ----
#pragma unroll
        for (int k = 0; k < TPB; ++k) {
            const int t = t0 + k;
            if (t < n_tokens) {
                const unsigned row = (unsigned)ids[t];
                v4f v = *reinterpret_cast<const v4f*>(weight + (size_t)row * D_DIM + off);
                __builtin_nontemporal_store(
                    v, reinterpret_cast<v4f*>(out + (size_t)t * D_DIM + off));
            }
        }
    }
}

extern "C" void kernel_launch(void* const* d_in, const int* in_sizes, int n_in,
                              void* d_out, int out_size, void* d_ws, size_t ws_size,
                              hipStream_t stream) {
    const int*   ids    = (const int*)d_in[0];     // token_ids, int32, B*S elems
    const float* weight = (const float*)d_in[1];   // [VOCAB, D] fp32
    float*       out    = (float*)d_out;           // [B*S, D] fp32

    const int n_tokens = in_sizes[0];              // 8192
    const int grid     = (n_tokens + TPB - 1) / TPB;  // 2048 blocks

    embed_gather_kernel<<<grid, 256, 0, stream>>>(ids, weight, out, n_tokens);
}